// SimpleGCN_7550552507130
// MI455X (gfx1250) — compile-verified
//
#include <hip/hip_runtime.h>

// Problem constants (match reference)
#define NN 50000   // nodes  (50000 = 3125 * 16, exact M-tiling)
#define NE 800000  // edges
#define CH 128     // IN_CH == HID == 128 (8 N-tiles of 16, K=128)
#define NG 512     // graphs

typedef float v2f __attribute__((ext_vector_type(2)));
typedef float v8f __attribute__((ext_vector_type(8)));

__device__ __forceinline__ void atomAddF(float* p, float v) {
  // lowers to global_atomic_add_f32 (no CAS loop)
  unsafeAtomicAdd(p, v);
}

// ---- degree: deg[d] += 1 per edge (self-loop added in k_dinv) ----
__global__ __launch_bounds__(256) void k_deg(const int* __restrict__ dst,
                                             float* __restrict__ deg) {
  int e = blockIdx.x * 256 + threadIdx.x;
  if (e < NE) atomAddF(&deg[dst[e]], 1.0f);
}

__global__ __launch_bounds__(256) void k_dinv(float* __restrict__ deg) {
  int n = blockIdx.x * 256 + threadIdx.x;
  if (n < NN) deg[n] = rsqrtf(deg[n] + 1.0f);  // +1 self loop => always > 0
}

// ---- fp32 GEMM via V_WMMA_F32_16X16X4_F32: C[M,128] = A[M,128] @ W[128,128] ----
// One wave per 16x16 output tile; block = 8 waves = one M-tile x all 8 N-tiles.
__global__ __launch_bounds__(256) void k_gemm(const float* __restrict__ A,
                                              const float* __restrict__ W,
                                              float* __restrict__ C) {
  const int nt   = threadIdx.x >> 5;  // N tile = wave id (0..7)
  const int lane = threadIdx.x & 31;
  const int mt   = blockIdx.x;        // M tile (0..3124)
  const int half = lane >> 4;         // lane group (K split)
  const int l16  = lane & 15;
  const float* arow = A + (size_t)(mt * 16 + l16) * CH;  // A row for this lane
  const int colB = nt * 16 + l16;                        // B column / C column
  v8f acc = {};
#pragma unroll
  for (int k0 = 0; k0 < CH; k0 += 4) {
    // A 16x4 fp32: lanes 0-15 regs hold K=k0+0,k0+1 ; lanes 16-31 hold K=k0+2,k0+3
    v2f a = *(const v2f*)(arow + k0 + 2 * half);
    // B 4x16 fp32: reg r holds row K = k0 + r + 2*half, col = lane%16
    v2f b;
    b.x = W[(k0 + 2 * half + 0) * CH + colB];
    b.y = W[(k0 + 2 * half + 1) * CH + colB];
    acc = __builtin_amdgcn_wmma_f32_16x16x4_f32(false, a, false, b,
                                                (short)0, acc, false, false);
  }
  // C/D layout: reg r -> row = r + 8*half, col = lane%16
#pragma unroll
  for (int r = 0; r < 8; ++r)
    C[(size_t)(mt * 16 + r + 8 * half) * CH + colB] = acc[r];
}

// ---- init aggregation buffer with the self-loop term: agg = xw * dinv[n]^2 ----
__global__ __launch_bounds__(256) void k_selfloop(const float* __restrict__ xw,
                                                  const float* __restrict__ dinv,
                                                  float* __restrict__ agg) {
  size_t i = (size_t)blockIdx.x * 256 + threadIdx.x;  // grid sized exactly NN*CH/256
  int n = (int)(i >> 7);
  float w = dinv[n];
  agg[i] = xw[i] * (w * w);
}

// ---- edge scatter-add: one wave per edge, float4 per lane ----
__global__ __launch_bounds__(256) void k_edge(const int* __restrict__ src,
                                              const int* __restrict__ dst,
                                              const float* __restrict__ dinv,
                                              const float* __restrict__ xw,
                                              float* __restrict__ agg) {
  int e = blockIdx.x * 8 + (threadIdx.x >> 5);
  if (e >= NE) return;
  int lane = threadIdx.x & 31;
  int s = src[e], d = dst[e];
  float w = dinv[s] * dinv[d];
  const float4 v = *(const float4*)(xw + (size_t)s * CH + lane * 4);
  float* o = agg + (size_t)d * CH + lane * 4;
  atomAddF(o + 0, v.x * w);
  atomAddF(o + 1, v.y * w);
  atomAddF(o + 2, v.z * w);
  atomAddF(o + 3, v.w * w);
}

__global__ __launch_bounds__(256) void k_bias_relu(float* __restrict__ h,
                                                   const float* __restrict__ b) {
  size_t i = (size_t)blockIdx.x * 256 + threadIdx.x;  // grid = NN*CH/256 exactly
  float v = h[i] + b[i & (CH - 1)];
  h[i] = v > 0.0f ? v : 0.0f;
}

// ---- mean pool accumulation: one block (128 threads) per node ----
__global__ __launch_bounds__(128) void k_pool(const float* __restrict__ h,
                                              const int* __restrict__ batch,
                                              float* __restrict__ pooled,
                                              float* __restrict__ cnt) {
  int n = blockIdx.x;
  int c = threadIdx.x;
  int g = batch[n];
  atomAddF(&pooled[(size_t)g * CH + c], h[(size_t)n * CH + c]);
  if (c == 0) atomAddF(&cnt[g], 1.0f);
}

// ---- out[g] = (pooled[g]/max(cnt,1)) . Wfc + bfc ----
__global__ __launch_bounds__(256) void k_final(const float* __restrict__ pooled,
                                               const float* __restrict__ cnt,
                                               const float* __restrict__ Wfc,
                                               const float* __restrict__ bfc,
                                               float* __restrict__ out) {
  int g = blockIdx.x * 256 + threadIdx.x;
  if (g >= NG) return;
  float inv = 1.0f / fmaxf(cnt[g], 1.0f);
  float acc = 0.0f;
#pragma unroll 8
  for (int c = 0; c < CH; ++c) acc += pooled[(size_t)g * CH + c] * Wfc[c];
  out[g] = acc * inv + bfc[0];
}

extern "C" void kernel_launch(void* const* d_in, const int* in_sizes, int n_in,
                              void* d_out, int out_size, void* d_ws, size_t ws_size,
                              hipStream_t stream) {
  const float* x   = (const float*)d_in[0];
  const int*   ei  = (const int*)d_in[1];   // [2, NE]: row 0 = src, row 1 = dst
  const int*   bat = (const int*)d_in[2];
  const float* W1  = (const float*)d_in[3];
  const float* b1  = (const float*)d_in[4];
  const float* W2  = (const float*)d_in[5];
  const float* b2  = (const float*)d_in[6];
  const float* Wfc = (const float*)d_in[7];
  const float* bfc = (const float*)d_in[8];
  float* out = (float*)d_out;
  const int* src = ei;
  const int* dst = ei + NE;

  // Workspace layout (256B-aligned slabs), ~49.3 MB total
  char* ws = (char*)d_ws;
  size_t off = 0;
  auto alloc = [&](size_t bytes) {
    void* p = ws + off;
    off += (bytes + 255) & ~(size_t)255;
    return p;
  };
  float* dinv   = (float*)alloc((size_t)NN * sizeof(float));       // deg -> dinv (in place)
  float* bufA   = (float*)alloc((size_t)NN * CH * sizeof(float));  // x@W
  float* bufB   = (float*)alloc((size_t)NN * CH * sizeof(float));  // aggregated / h
  float* pooled = (float*)alloc((size_t)NG * CH * sizeof(float));  // contiguous with cnt
  float* cnt    = (float*)alloc((size_t)NG * sizeof(float));

  const int nEdgeBlk = (NE + 255) / 256;   // 3125
  const int nNodeBlk = (NN + 255) / 256;   // 196
  const int nFeatBlk = (NN * CH) / 256;    // 25000 (exact)
  const int nWaveEdg = (NE + 7) / 8;       // 100000 (8 waves / block)

  // degrees -> dinv
  hipMemsetAsync(dinv, 0, (size_t)NN * sizeof(float), stream);
  k_deg<<<nEdgeBlk, 256, 0, stream>>>(dst, dinv);
  k_dinv<<<nNodeBlk, 256, 0, stream>>>(dinv);

  // ---- layer 1 ----
  k_gemm<<<NN / 16, 256, 0, stream>>>(x, W1, bufA);
  k_selfloop<<<nFeatBlk, 256, 0, stream>>>(bufA, dinv, bufB);
  k_edge<<<nWaveEdg, 256, 0, stream>>>(src, dst, dinv, bufA, bufB);
  k_bias_relu<<<nFeatBlk, 256, 0, stream>>>(bufB, b1);

  // ---- layer 2 ----
  k_gemm<<<NN / 16, 256, 0, stream>>>(bufB, W2, bufA);
  k_selfloop<<<nFeatBlk, 256, 0, stream>>>(bufA, dinv, bufB);
  k_edge<<<nWaveEdg, 256, 0, stream>>>(src, dst, dinv, bufA, bufB);
  k_bias_relu<<<nFeatBlk, 256, 0, stream>>>(bufB, b2);

  // ---- pool + fc ----
  hipMemsetAsync(pooled, 0, ((size_t)NG * CH + NG) * sizeof(float), stream);
  k_pool<<<NN, CH, 0, stream>>>(bufB, bat, pooled, cnt);
  k_final<<<(NG + 255) / 256, 256, 0, stream>>>(pooled, cnt, Wfc, bfc, out);
}